// WarpAdjoint_31069793419523
// MI455X (gfx1250) — compile-verified
//
#include <hip/hip_runtime.h>
#include <stdint.h>

// Problem: adjoint bilinear warp-splat, B=8, C=12, M=N=512.
//   out[b,y,x] += sum_c sum_corners w * x[b,c,i,j]  at (i+uy, j+ux) corners.
// Memory-bound: ~310 MB traffic -> ~13us at 23.3 TB/s HBM.
// Strategy (CDNA5):
//   * TDM tensor_load_to_lds streams 2D x/u tiles into LDS, double-buffered,
//     synchronized with s_wait_tensorcnt (TENSORcnt).
//   * 48x48 LDS accumulator tile (halo 8) absorbs the splat via ds_add_f32
//     atomics; 12 channels reduced on-chip; single atomic flush to the
//     L2-resident 8.4MB output. Out-of-halo targets (P ~ 1e-15 for N(0,1)
//     displacements) fall back to global atomicAdd for exactness.

#define Bsz   8
#define Csz   12
#define Msz   512
#define Nsz   512
#define TILE  32
#define HALO  8
#define ACCW  (TILE + 2 * HALO)   // 48
#define ACCSZ (ACCW * ACCW)       // 2304
#define NTHREADS 256

typedef unsigned int v4u __attribute__((ext_vector_type(4)));
typedef int          v8i __attribute__((ext_vector_type(8)));
typedef int          v4i __attribute__((ext_vector_type(4)));

// Generic pointer to an LDS object: low 32 bits are the LDS byte offset
// (addrspacecast local->flat keeps the group-segment offset in addr[31:0]).
__device__ __forceinline__ uint32_t lds_byte_off(const void* p) {
  return (uint32_t)(uintptr_t)p;
}

// Build a 2D TDM descriptor (D#) and issue TENSOR_LOAD_TO_LDS.
// Tile: tile_h rows of tile_w 4-byte elements, row pitch = stride_elems.
// D# layout per CDNA5 ISA (groups 2/3 zero for 2D).
// This toolchain exposes the 6-arg builtin: (v4u, v8i, v4i, v4i, v8i, cpol);
// the trailing v8i has no architectural D# group -> zeros.
__device__ __forceinline__ void tdm_load_2d(uint32_t lds_addr,
                                            const void* gptr,
                                            uint32_t tile_w,
                                            uint32_t tile_h,
                                            uint32_t stride_elems) {
  uint64_t ga = (uint64_t)(uintptr_t)gptr;
  v4u g0;
  g0.x = 1u;                                            // count=1 (valid), user mode
  g0.y = lds_addr;                                      // lds_addr (bytes)
  g0.z = (uint32_t)(ga & 0xFFFFFFFFu);                  // global_addr[31:0]
  g0.w = (uint32_t)((ga >> 32) & 0x01FFFFFFu)           // global_addr[56:32]
       | (2u << 30);                                    // type=2 ("image")
  v8i g1;
  g1[0] = (int)(2u << 16);                              // data_size=2 -> 4 bytes
  g1[1] = (int)((tile_w & 0xFFFFu) << 16);              // tensor_dim0[15:0]
  g1[2] = (int)(((tile_w >> 16) & 0xFFFFu)              // tensor_dim0[31:16]
       |        ((tile_h & 0xFFFFu) << 16));            // tensor_dim1[15:0]
  g1[3] = (int)(((tile_h >> 16) & 0xFFFFu)              // tensor_dim1[31:16]
       |        ((tile_w & 0xFFFFu) << 16));            // tile_dim0
  g1[4] = (int)(tile_h & 0xFFFFu);                      // tile_dim1 (tile_dim2=0)
  g1[5] = (int)stride_elems;                            // tensor_dim0_stride[31:0]
  g1[6] = 0;                                            // stride hi / dim1_stride lo
  g1[7] = 0;
  v4i g2 = {0, 0, 0, 0};                                // 2D: dims 2..4 unused
  v4i g3 = {0, 0, 0, 0};
  v8i g4 = {0, 0, 0, 0, 0, 0, 0, 0};                    // no architectural group
  __builtin_amdgcn_tensor_load_to_lds(g0, g1, g2, g3, g4, 0 /*cpol*/);
}

__global__ __launch_bounds__(NTHREADS)
void WarpAdjoint_31069793419523_kernel(const float* __restrict__ x,
                                       const float* __restrict__ u,
                                       float* __restrict__ out) {
  __shared__ float xs[2][TILE * TILE];        //  8 KB staging (x), double buffer
  __shared__ float us[2][TILE * TILE * 2];    // 16 KB staging (u), double buffer
  __shared__ float acc[ACCSZ];                //  9 KB accumulator tile (+halo)

  const int tid = threadIdx.x;
  const int tiles_x = Nsz / TILE;             // 16
  const int tiles_y = Msz / TILE;             // 16
  const int bidx = blockIdx.x;
  const int tx = bidx % tiles_x;
  const int ty = (bidx / tiles_x) % tiles_y;
  const int b  = bidx / (tiles_x * tiles_y);
  const int i0 = ty * TILE;
  const int j0 = tx * TILE;

  const bool issuer = (tid < 32);   // wave 0 only issues TDM (branch-level guard)

  auto issue = [&](int buf, int c) {
    const float* xg = x + (((size_t)b * Csz + c) * Msz + i0) * Nsz + j0;
    const float* ug = u + (((((size_t)b * Csz + c) * Msz + i0) * Nsz + j0) << 1);
    tdm_load_2d(lds_byte_off(&xs[buf][0]), xg, TILE,     TILE, Nsz);
    tdm_load_2d(lds_byte_off(&us[buf][0]), ug, TILE * 2, TILE, Nsz * 2);
  };

  // Kick off channel 0's DMA first so it overlaps the accumulator clear.
  if (issuer) issue(0, 0);

  for (int k = tid; k < ACCSZ; k += NTHREADS) acc[k] = 0.0f;

  if (issuer) __builtin_amdgcn_s_wait_tensorcnt(0);
  __syncthreads();

  auto scatter = [&](int yc, int xc, float v) {
    if ((unsigned)yc < (unsigned)Msz && (unsigned)xc < (unsigned)Nsz) {
      int ly = yc - i0 + HALO;
      int lx = xc - j0 + HALO;
      if ((unsigned)ly < (unsigned)ACCW && (unsigned)lx < (unsigned)ACCW) {
        atomicAdd(&acc[ly * ACCW + lx], v);               // ds_add_f32
      } else {
        atomicAdd(&out[((size_t)b * Msz + yc) * Nsz + xc], v);  // rare fallback
      }
    }
  };

  for (int c = 0; c < Csz; ++c) {
    const int cur = c & 1;
    // Prefetch next channel's tiles into the other buffer (overlaps compute).
    if (issuer && (c + 1) < Csz) issue(cur ^ 1, c + 1);

    #pragma unroll
    for (int pp = 0; pp < (TILE * TILE) / NTHREADS; ++pp) {
      const int p  = tid + pp * NTHREADS;     // 0..1023
      const int li = p >> 5;
      const int lj = p & (TILE - 1);
      const float xv = xs[cur][p];
      const float ux = us[cur][2 * p + 0];
      const float uy = us[cur][2 * p + 1];
      const float px = (float)(j0 + lj) + ux;
      const float py = (float)(i0 + li) + uy;
      const float x0f = floorf(px);
      const float y0f = floorf(py);
      const float wx = px - x0f;
      const float wy = py - y0f;
      const int x0 = (int)x0f;
      const int y0 = (int)y0f;
      const float omwx = 1.0f - wx;
      const float omwy = 1.0f - wy;
      scatter(y0,     x0,     omwx * omwy * xv);
      scatter(y0,     x0 + 1, wx   * omwy * xv);
      scatter(y0 + 1, x0,     omwx * wy   * xv);
      scatter(y0 + 1, x0 + 1, wx   * wy   * xv);
    }
    __syncthreads();                          // all reads of buf[cur] done
    if (issuer && (c + 1) < Csz) __builtin_amdgcn_s_wait_tensorcnt(0);
    __syncthreads();                          // buf[cur^1] now valid for c+1
  }

  // Flush accumulator tile (incl. halo) with global f32 atomics (L2-resident).
  for (int k = tid; k < ACCSZ; k += NTHREADS) {
    const float v = acc[k];
    const int ly = k / ACCW;
    const int lx = k - ly * ACCW;
    const int gy = i0 + ly - HALO;
    const int gx = j0 + lx - HALO;
    if (v != 0.0f && (unsigned)gy < (unsigned)Msz && (unsigned)gx < (unsigned)Nsz)
      atomicAdd(&out[((size_t)b * Msz + gy) * Nsz + gx], v);
  }
}

__global__ void WarpAdjoint_zero_kernel(float* __restrict__ out, int n) {
  int i = blockIdx.x * blockDim.x + threadIdx.x;
  if (i < n) out[i] = 0.0f;
}

extern "C" void kernel_launch(void* const* d_in, const int* in_sizes, int n_in,
                              void* d_out, int out_size, void* d_ws, size_t ws_size,
                              hipStream_t stream) {
  (void)in_sizes; (void)n_in; (void)d_ws; (void)ws_size;
  const float* x = (const float*)d_in[0];
  const float* u = (const float*)d_in[1];
  float* out = (float*)d_out;

  WarpAdjoint_zero_kernel<<<(out_size + NTHREADS - 1) / NTHREADS, NTHREADS, 0, stream>>>(
      out, out_size);

  const int nblocks = Bsz * (Msz / TILE) * (Nsz / TILE);  // 2048
  WarpAdjoint_31069793419523_kernel<<<nblocks, NTHREADS, 0, stream>>>(x, u, out);
}